// Simulator_19370302505658
// MI455X (gfx1250) — compile-verified
//
#include <hip/hip_runtime.h>
#include <math.h>

typedef __attribute__((ext_vector_type(2))) float v2f;
typedef __attribute__((ext_vector_type(8))) float v8f;

#define N_OBJ   8192
#define N_FLOOR 8192
#define TILES   512          // 8192 / 16
#define RAD     0.001f
#define DT_     0.01f
#define STEPS_  100
#define SLEEP_TH_ (DT_ * 9.8f)

// ---------------------------------------------------------------------------
// Kernel 1: pack floor particles into the exact B-matrix (4x16 f32) lane/VGPR
// layout expected by V_WMMA_F32_16X16X4_F32.
//   Per tile ft (16 floor points), per lane L (N = L&15), per vgpr v:
//     lanes  0-15: V0 = K0 = -2*fx,  V1 = K1 = -2*fy
//     lanes 16-31: V0 = K2 =  1.0,   V1 = K3 = fx^2+fy^2
//   Stored as bpack[ft*64 + lane*2 + v]  -> one b64 load per lane per tile.
// ---------------------------------------------------------------------------
__global__ void pack_b_kernel(const float* __restrict__ floor_pc,
                              float* __restrict__ bpack) {
  int idx = blockIdx.x * blockDim.x + threadIdx.x;    // TILES*64 entries
  if (idx >= TILES * 64) return;
  int ft   = idx >> 6;
  int s    = idx & 63;
  int lane = s >> 1;
  int v    = s & 1;
  int j    = ft * 16 + (lane & 15);
  float fx = floor_pc[j * 3 + 0];
  float fy = floor_pc[j * 3 + 1];
  float val;
  if (lane < 16) val = (v == 0) ? (-2.0f * fx) : (-2.0f * fy);
  else           val = (v == 0) ? 1.0f : (fx * fx + fy * fy);
  bpack[idx] = val;
}

// ---------------------------------------------------------------------------
// Kernel 2: per object tile (16 rows), min over all floor tiles of
//   dxy2 = |o|^2 + |f|^2 - 2 o.f   via  V_WMMA_F32_16X16X4_F32.
// A layout (16x4 f32): lanes 0-15 -> (K0=ox, K1=oy); lanes 16-31 -> (K2=|o|^2, K3=1).
// One wave per tile; 8 waves (256 threads) per workgroup.
// ---------------------------------------------------------------------------
__global__ void __launch_bounds__(256)
minxy_kernel(const float* __restrict__ obj_pc,
             const float* __restrict__ bpack,
             float* __restrict__ minxy2) {
  const int wave = blockIdx.x * (blockDim.x >> 5) + (threadIdx.x >> 5);
  const int lane = threadIdx.x & 31;
  const int m    = lane & 15;
  const int oi   = wave * 16 + m;

  float ox = obj_pc[oi * 3 + 0];
  float oy = obj_pc[oi * 3 + 1];
  bool  lo = lane < 16;
  v2f a;
  a.x = lo ? ox : (ox * ox + oy * oy);
  a.y = lo ? oy : 1.0f;

  v8f rmin;
#pragma unroll
  for (int k = 0; k < 8; ++k) rmin[k] = 3.402823466e38f;

  const v2f* bp = ((const v2f*)bpack) + lane;   // + ft*32 per tile
#pragma unroll 4
  for (int ft = 0; ft < TILES; ++ft) {
    v2f b = bp[ft * 32];
    v8f c = {0.f, 0.f, 0.f, 0.f, 0.f, 0.f, 0.f, 0.f};
    // 8 args: (neg_a, A, neg_b, B, c_mod, C, reuse_a, reuse_b)
    c = __builtin_amdgcn_wmma_f32_16x16x4_f32(
        false, a, false, b, (short)0, c, false, false);
#pragma unroll
    for (int k = 0; k < 8; ++k) rmin[k] = fminf(rmin[k], c[k]);
  }

  // Reduce across N within each 16-lane half (rows 0-7 in lanes 0-15,
  // rows 8-15 in lanes 16-31).
#pragma unroll
  for (int mask = 1; mask < 16; mask <<= 1) {
#pragma unroll
    for (int k = 0; k < 8; ++k)
      rmin[k] = fminf(rmin[k], __shfl_xor(rmin[k], mask, 32));
  }
  if ((lane & 15) == 0) {
    int base = wave * 16 + (lane >> 4) * 8;
#pragma unroll
    for (int k = 0; k < 8; ++k) minxy2[base + k] = rmin[k];
  }
}

// ---------------------------------------------------------------------------
// Kernel 3: 100-step rigid drop; translation is purely vertical.
// Single block of 1024 threads, 8 particles per thread, LDS max-reduction
// per step for the global penetration projection.
// ---------------------------------------------------------------------------
__global__ void __launch_bounds__(1024)
sim_kernel(const float* __restrict__ obj_pc,
           const float* __restrict__ minxy2,
           float* __restrict__ out_p,      // [8192*3]
           float* __restrict__ out_mask) { // [8192]
  __shared__ float red[32];
  __shared__ float bc;
  const int tid = threadIdx.x;
  const int P = N_OBJ / 1024;   // 8

  float mz[P], oz[P], tzf[P];
  bool  col[P];
#pragma unroll
  for (int k = 0; k < P; ++k) {
    int idx = tid + k * 1024;
    mz[k]  = minxy2[idx];
    oz[k]  = obj_pc[idx * 3 + 2];
    col[k] = false;
    tzf[k] = 0.0f;                // p_first defaults to obj_pc
  }

  float tz = 0.0f, vz = 0.0f;
  bool asleep = false;

  for (int s = 0; s < STEPS_; ++s) {
    if (asleep) break;            // uniform across block (scalar state)
    float vn = vz - 9.8f * DT_;
    float tn = tz + vn * DT_;

    float pen[P];
    float lmax = 0.0f;            // max over where(contact, pen, 0)
#pragma unroll
    for (int k = 0; k < P; ++k) {
      float z = oz[k] + tn;
      float d = sqrtf(mz[k] + z * z + 1e-12f);
      pen[k] = 2.0f * RAD - d;
      lmax = fmaxf(lmax, pen[k]);
    }
    // wave reduce
#pragma unroll
    for (int mask = 16; mask >= 1; mask >>= 1)
      lmax = fmaxf(lmax, __shfl_xor(lmax, mask, 32));
    if ((tid & 31) == 0) red[tid >> 5] = lmax;
    __syncthreads();
    if (tid < 32) {
      float v = red[tid];
#pragma unroll
      for (int mask = 16; mask >= 1; mask >>= 1)
        v = fmaxf(v, __shfl_xor(v, mask, 32));
      if (tid == 0) bc = v;
    }
    __syncthreads();
    float maxpen = bc;
    __syncthreads();              // protect bc/red before next iteration

    bool  any_c = maxpen > 0.0f;
    float t_c = tn + (any_c ? maxpen : 0.0f);
    float v_c = any_c ? 0.0f : vn;     // xy velocity is always zero
#pragma unroll
    for (int k = 0; k < P; ++k) {
      bool contact = pen[k] > 0.0f;
      if (contact && !col[k]) tzf[k] = t_c;
      col[k] = col[k] || contact;
    }
    asleep = asleep || (any_c && (fabsf(v_c) < SLEEP_TH_));
    tz = t_c;
    vz = v_c;
  }

#pragma unroll
  for (int k = 0; k < P; ++k) {
    int idx = tid + k * 1024;
    out_p[idx * 3 + 0] = obj_pc[idx * 3 + 0];
    out_p[idx * 3 + 1] = obj_pc[idx * 3 + 1];
    out_p[idx * 3 + 2] = oz[k] + tzf[k];
    out_mask[idx] = col[k] ? 1.0f : 0.0f;
  }
}

// ---------------------------------------------------------------------------
extern "C" void kernel_launch(void* const* d_in, const int* in_sizes, int n_in,
                              void* d_out, int out_size, void* d_ws, size_t ws_size,
                              hipStream_t stream) {
  const float* obj_pc   = (const float*)d_in[0];   // [8192,3]
  const float* floor_pc = (const float*)d_in[1];   // [8192,3]
  // d_in[2] = iter_step, d_in[3] = obj_idx : unused by the reference output

  float* bpack  = (float*)d_ws;                              // 512*64 floats
  float* minxy2 = (float*)((char*)d_ws + TILES * 64 * sizeof(float)); // 8192 floats

  float* out_p    = (float*)d_out;          // 24576 floats
  float* out_mask = out_p + N_OBJ * 3;      // 8192 floats (bool as 0/1)

  pack_b_kernel<<<(TILES * 64 + 255) / 256, 256, 0, stream>>>(floor_pc, bpack);
  minxy_kernel<<<TILES / 8, 256, 0, stream>>>(obj_pc, bpack, minxy2);
  sim_kernel<<<1, 1024, 0, stream>>>(obj_pc, minxy2, out_p, out_mask);
}